// YOLOv5ObjectDetector_32736240730786
// MI455X (gfx1250) — compile-verified
//
#include <hip/hip_runtime.h>

// ---------------- problem constants (match reference) ----------------
constexpr int   B_      = 16;
constexpr int   N_      = 25200;
constexpr int   NC_     = 80;
constexpr int   K_      = 2048;
constexpr int   MAXDET_ = 300;
constexpr float CONF_   = 0.4f;
constexpr float IOUTH_  = 0.45f;
constexpr float MAXWH_  = 4096.0f;
constexpr int   NT_     = (N_ + K_ - 1) / K_;   // 13 tiles

// rows are 85 floats; a 32-row tile = 2720 dwords = 680 b128 chunks, 16B aligned
constexpr int   ROWF_    = 85;
constexpr int   TILEDW_  = 32 * ROWF_;           // 2720
constexpr int   TILEB128_= TILEDW_ / 4;          // 680
constexpr int   NTILES_  = (B_ * N_) / 32;       // 12600 (exact)
constexpr int   WPB_     = 4;                    // waves per block in score kernel

// ---------------- CDNA5 async global->LDS copy (gfx1250) ----------------
// global_load_async_to_lds_b128: VDST = VGPR holding LDS byte address,
// VADDR = VGPR pair holding 64-bit global address. Tracked by ASYNCcnt.
__device__ __forceinline__ void async_copy_b128(unsigned lds_addr, const void* gaddr) {
    asm volatile("global_load_async_to_lds_b128 %0, %1, off"
                 :: "v"(lds_addr), "v"(gaddr)
                 : "memory");
}
__device__ __forceinline__ void wait_async0() {
    asm volatile("s_wait_asynccnt 0" ::: "memory");
}

// ---------------- sortable-float key helpers ----------------
__device__ __forceinline__ unsigned long long make_key(float s, int n) {
    unsigned u = __float_as_uint(s);
    u = (u & 0x80000000u) ? ~u : (u | 0x80000000u);
    return ((unsigned long long)u << 32) | (unsigned)(~n);
}
__device__ __forceinline__ float key_score(unsigned long long k) {
    unsigned hi = (unsigned)(k >> 32);
    unsigned u = (hi & 0x80000000u) ? (hi ^ 0x80000000u) : ~hi;
    return __uint_as_float(u);
}

// ---------------- K1: per-row confidence score (async-staged, coalesced) ----------------
// One wave = one 32-row tile. The wave streams its 10,880B tile into LDS with
// coalesced async b128 transfers, waits on its own ASYNCcnt (no block barrier
// needed), then each lane scores its row out of LDS (bank-conflict-free).
__global__ __launch_bounds__(WPB_ * 32)
void score_kernel(const float* __restrict__ pred, float* __restrict__ scores) {
    __shared__ float srows[WPB_][TILEDW_];
    const int wave = threadIdx.x >> 5;
    const int lane = threadIdx.x & 31;
    const int tile = blockIdx.x * WPB_ + wave;      // grid sized exactly

    const float* src = pred + (size_t)tile * TILEDW_;
#pragma unroll
    for (int it = 0; it < (TILEB128_ + 31) / 32; ++it) {
        int chunk = it * 32 + lane;
        if (chunk < TILEB128_) {
            unsigned la = (unsigned)(unsigned long long)(const void*)&srows[wave][chunk * 4];
            async_copy_b128(la, src + chunk * 4);
        }
    }
    wait_async0();          // own-wave counter: staged data now visible in LDS

    const float* row = &srows[wave][lane * ROWF_];
    float obj = row[4];
    float best = -1e30f;
#pragma unroll 5
    for (int c = 0; c < NC_; ++c) best = fmaxf(best, row[5 + c] * obj);
    bool valid = (obj > CONF_) && (best > CONF_);
    scores[tile * 32 + lane] = valid ? best : -1.0f;
}

// ---------------- bitonic helpers (2048 keys, 1024 threads) ----------------
__device__ __forceinline__ void bitonic_sort_desc(unsigned long long* key, int tid) {
    for (int k = 2; k <= K_; k <<= 1) {
        for (int j = k >> 1; j > 0; j >>= 1) {
            __syncthreads();
#pragma unroll
            for (int s = 0; s < 2; ++s) {
                int i = tid + s * 1024;
                int ixj = i ^ j;
                if (ixj > i) {
                    unsigned long long a = key[i], b = key[ixj];
                    bool descSeg = ((i & k) == 0);
                    if (descSeg ? (a < b) : (a > b)) { key[i] = b; key[ixj] = a; }
                }
            }
        }
    }
    __syncthreads();
}
__device__ __forceinline__ void bitonic_merge_desc(unsigned long long* key, int tid) {
    for (int j = K_ >> 1; j > 0; j >>= 1) {
        __syncthreads();
#pragma unroll
        for (int s = 0; s < 2; ++s) {
            int i = tid + s * 1024;
            int ixj = i ^ j;
            if (ixj > i) {
                unsigned long long a = key[i], b = key[ixj];
                if (a < b) { key[i] = b; key[ixj] = a; }
            }
        }
    }
    __syncthreads();
}
__device__ __forceinline__ void load_tile(unsigned long long* key, const float* sc,
                                          int tile, int tid) {
#pragma unroll
    for (int s = 0; s < 2; ++s) {
        int e = tid + s * 1024;
        int n = tile * K_ + e;
        unsigned long long k = 0ull;
        if (n < N_) k = make_key(sc[n], n);
        key[e] = k;
    }
}

// ---------------- K2: top-2048 select + gather per image ----------------
__global__ __launch_bounds__(1024)
void topk_kernel(const float* __restrict__ scores, const float* __restrict__ pred,
                 float* __restrict__ topsc, int* __restrict__ topidx,
                 float* __restrict__ clsv, float4* __restrict__ box,
                 float4* __restrict__ offbox) {
    __shared__ unsigned long long keyA[K_];
    __shared__ unsigned long long keyB[K_];
    const int b = blockIdx.x, tid = threadIdx.x;
    const float* sc = scores + (size_t)b * N_;

    load_tile(keyA, sc, 0, tid);
    bitonic_sort_desc(keyA, tid);
    for (int t = 1; t < NT_; ++t) {
        load_tile(keyB, sc, t, tid);
        bitonic_sort_desc(keyB, tid);
        // keep top-2048 of (A desc) U (B desc): elementwise max vs reversed B -> bitonic
#pragma unroll
        for (int s = 0; s < 2; ++s) {
            int i = tid + s * 1024;
            unsigned long long a = keyA[i], bb = keyB[(K_ - 1) - i];
            keyA[i] = (a > bb) ? a : bb;
        }
        bitonic_merge_desc(keyA, tid);
    }

    // gather: decode, recompute box / class for the selected rows
#pragma unroll
    for (int s = 0; s < 2; ++s) {
        int k = tid + s * 1024;
        unsigned long long kv = keyA[k];
        float scv; int idx;
        if (kv == 0ull) { scv = -1.0f; idx = 0; }
        else { idx = (int)~(unsigned)kv; scv = key_score(kv); }
        const float* row = pred + ((size_t)b * N_ + idx) * 85;
        float cx = row[0], cy = row[1], w = row[2], h = row[3], obj = row[4];
        float best = -1e30f; int bj = 0;
        for (int c = 0; c < NC_; ++c) {
            float v = row[5 + c] * obj;
            if (v > best) { best = v; bj = c; }
        }
        float clsf = (float)bj;
        float x1 = cx - w * 0.5f, y1 = cy - h * 0.5f;
        float x2 = cx + w * 0.5f, y2 = cy + h * 0.5f;
        float off = clsf * MAXWH_;
        int bi = b * K_ + k;
        topsc[bi]  = scv;
        topidx[bi] = idx;
        clsv[bi]   = clsf;
        box[bi]    = make_float4(x1, y1, x2, y2);
        offbox[bi] = make_float4(x1 + off, y1 + off, x2 + off, y2 + off);
    }
}

// ---------------- K3: suppression-bitmask build (async LDS staging) ----------------
__global__ __launch_bounds__(256)
void iou_mask_kernel(const float4* __restrict__ offbox, unsigned* __restrict__ mask) {
    __shared__ float4 sbox[K_];
    __shared__ float  sarea[K_];
    const int tid = threadIdx.x;
    const int b = blockIdx.x >> 3;          // 8 blocks per image
    const int itile = blockIdx.x & 7;

    const float4* src = offbox + (size_t)b * K_;
#pragma unroll
    for (int q = 0; q < 8; ++q) {           // 2048 b128 async transfers per block
        int k = tid + q * 256;
        unsigned la = (unsigned)(unsigned long long)(const void*)&sbox[k];
        async_copy_b128(la, src + k);
    }
    wait_async0();
    __syncthreads();
#pragma unroll
    for (int q = 0; q < 8; ++q) {
        int k = tid + q * 256;
        float4 v = sbox[k];
        sarea[k] = (v.z - v.x) * (v.w - v.y);
    }
    __syncthreads();

    const int i = itile * 256 + tid;
    const float4 bi = sbox[i];
    const float  ai = sarea[i];
    unsigned* mrow = mask + ((size_t)(b * K_ + i)) * 64;
    for (int w = 0; w < 64; ++w) {
        unsigned bits = 0u;
        int jbase = w * 32;
        if (jbase + 31 > i) {
            for (int t = 0; t < 32; ++t) {
                int j = jbase + t;
                if (j > i) {
                    float4 bj = sbox[j];
                    float lx = fmaxf(bi.x, bj.x), ly = fmaxf(bi.y, bj.y);
                    float rx = fminf(bi.z, bj.z), ry = fminf(bi.w, bj.w);
                    float iw = fmaxf(rx - lx, 0.0f), ih = fmaxf(ry - ly, 0.0f);
                    float inter = iw * ih;
                    float iou = inter / (ai + sarea[j] - inter + 1e-7f);
                    if (iou > IOUTH_) bits |= (1u << t);
                }
            }
        }
        mrow[w] = bits;
    }
}

// ---------------- K4: serial greedy reduce, one wave32 per image ----------------
__global__ __launch_bounds__(32)
void nms_reduce_kernel(const float* __restrict__ topsc, const unsigned* __restrict__ mask,
                       int* __restrict__ slotmap) {
    const int b = blockIdx.x, lane = threadIdx.x;
    const float* sc = topsc + (size_t)b * K_;
    const unsigned* mb = mask + (size_t)b * K_ * 64;
    unsigned r0 = 0u, r1 = 0u;                      // remv words lane, lane+32
    for (int i = 0; i < K_; ++i) {
        int wi = i >> 5;
        unsigned word = __shfl((wi < 32) ? r0 : r1, wi & 31, 32);
        bool alive = (sc[i] > 0.0f) && !((word >> (i & 31)) & 1u);
        if (alive) {
            const unsigned* mrow = mb + (size_t)i * 64;
            r0 |= mrow[lane];
            r1 |= mrow[lane + 32];
        }
    }
    int* sm = slotmap + b * MAXDET_;
    for (int r = lane; r < MAXDET_; r += 32) sm[r] = -1;
    int rank = 0;
    for (int w = 0; w < 64; ++w) {
        unsigned remw = __shfl((w < 32) ? r0 : r1, w & 31, 32);
        int i = w * 32 + lane;
        bool keep = (sc[i] > 0.0f) && !((remw >> lane) & 1u);
        unsigned km = (unsigned)__ballot(keep);
        int my = rank + __popc(km & ((1u << lane) - 1u));
        if (keep && my < MAXDET_) sm[my] = i;
        rank += __popc(km);
    }
}

// ---------------- K5: compaction + logits gather + zero fill ----------------
__global__ __launch_bounds__(96)
void output_kernel(const int* __restrict__ slotmap, const float4* __restrict__ box,
                   const float* __restrict__ topsc, const float* __restrict__ clsv,
                   const int* __restrict__ topidx, const float* __restrict__ logits,
                   float* __restrict__ out) {
    const int ro = blockIdx.x;           // 0 .. B*300-1
    const int c = threadIdx.x;
    if (c >= 86) return;
    const int b = ro / MAXDET_;
    float v = 0.0f;
    int i = slotmap[ro];
    if (i >= 0) {
        int bi = b * K_ + i;
        if (c < 4) {
            float4 bb = box[bi];
            v = (c == 0) ? bb.x : (c == 1) ? bb.y : (c == 2) ? bb.z : bb.w;
        } else if (c == 4) v = topsc[bi];
        else if (c == 5)   v = clsv[bi];
        else               v = logits[((size_t)b * N_ + topidx[bi]) * NC_ + (c - 6)];
    }
    out[(size_t)ro * 86 + c] = v;
}

// ---------------- host entry ----------------
extern "C" void kernel_launch(void* const* d_in, const int* in_sizes, int n_in,
                              void* d_out, int out_size, void* d_ws, size_t ws_size,
                              hipStream_t stream) {
    (void)in_sizes; (void)n_in; (void)out_size; (void)ws_size;
    const float* pred   = (const float*)d_in[0];
    const float* logits = (const float*)d_in[1];
    float* out = (float*)d_out;

    char* p = (char*)d_ws;
    auto carve = [&](size_t bytes) { void* r = (void*)p; p += (bytes + 255) & ~(size_t)255; return r; };
    float*    scores  = (float*)   carve((size_t)B_ * N_ * 4);
    float*    topsc   = (float*)   carve((size_t)B_ * K_ * 4);
    float*    clsv    = (float*)   carve((size_t)B_ * K_ * 4);
    int*      topidx  = (int*)     carve((size_t)B_ * K_ * 4);
    float4*   box     = (float4*)  carve((size_t)B_ * K_ * 16);
    float4*   offbox  = (float4*)  carve((size_t)B_ * K_ * 16);
    unsigned* mask    = (unsigned*)carve((size_t)B_ * K_ * 64 * 4);
    int*      slotmap = (int*)     carve((size_t)B_ * MAXDET_ * 4);

    score_kernel<<<NTILES_ / WPB_, WPB_ * 32, 0, stream>>>(pred, scores);
    topk_kernel<<<B_, 1024, 0, stream>>>(scores, pred, topsc, topidx, clsv, box, offbox);
    iou_mask_kernel<<<B_ * (K_ / 256), 256, 0, stream>>>(offbox, mask);
    nms_reduce_kernel<<<B_, 32, 0, stream>>>(topsc, mask, slotmap);
    output_kernel<<<B_ * MAXDET_, 96, 0, stream>>>(slotmap, box, topsc, clsv, topidx, logits, out);
}